// TrajectoryLoss_65652870087165
// MI455X (gfx1250) — compile-verified
//
#include <hip/hip_runtime.h>

// ---------------------------------------------------------------------------
// TrajectoryLoss for MI455X (gfx1250, wave32).
//
// Collision term mapped onto the matrix engine via the Gram-matrix trick:
//   d2(i,j) = |x_i|^2 + |x_j|^2 - 2 * (x_i . x_j)
// 128x128 inner-product matrix per (b,t) via V_WMMA_F32_16X16X4_F32 tiles
// (K=2 coords zero-padded to K=4). 36 upper-tri tiles, 9 per wave, fully
// unrolled. Tile (I,J) decode is forced scalar (readfirstlane + const
// table -> s_load), operand fetches are unconditional ds_load_b64 +
// v_cndmask (no exec-divergent loads). Masked-out agents are staged with
// huge sentinel coords so the epilogue needs no mask test, and the raw
// v_sqrt_f32 builtin avoids the precise-sqrt fixup expansion.
// ---------------------------------------------------------------------------

typedef __attribute__((ext_vector_type(2))) float v2f;
typedef __attribute__((ext_vector_type(8))) float v8f;

#define ADE_W    1.0f
#define FDE_W    1.0f
#define COLL_W   0.5f
#define COLL_THR 0.1f

static constexpr int BB = 32, TT = 50, AA = 128;

// Packed (I<<4)|J for the 36 upper-triangular tiles of the 8x8 tile grid.
__device__ const int kTileIJ[36] = {
    0x00, 0x01, 0x02, 0x03, 0x04, 0x05, 0x06, 0x07,
    0x11, 0x12, 0x13, 0x14, 0x15, 0x16, 0x17,
    0x22, 0x23, 0x24, 0x25, 0x26, 0x27,
    0x33, 0x34, 0x35, 0x36, 0x37,
    0x44, 0x45, 0x46, 0x47,
    0x55, 0x56, 0x57,
    0x66, 0x67,
    0x77};

// ws accumulators: [0]=collision penalty sum, [1]=collision count,
//                  [2]=masked distance sum (ADE numerator), [3]=FDE numerator

// ---------------------------------------------------------------------------
// Kernel 1: masked ADE/FDE partial sums. Two (b,t,a) elements per thread,
// b128 global loads.
// ---------------------------------------------------------------------------
__global__ void ade_fde_kernel(const float* __restrict__ pred,
                               const float* __restrict__ gt,
                               const unsigned char* __restrict__ mask,
                               float* __restrict__ acc) {
  const int tid = blockIdx.x * blockDim.x + threadIdx.x;  // < B*T*A/2
  float dsum = 0.0f, fsum = 0.0f;
  if (tid < BB * TT * AA / 2) {
    const int idx = tid * 2;          // first of two consecutive agents
    const int a   = idx % AA;         // even; a and a+1 share (b,t)
    const int bt  = idx / AA;
    const int t   = bt % TT;
    const int b   = bt / TT;
    float4 p = ((const float4*)pred)[tid];   // (x0,y0,x1,y1)
    float4 g = ((const float4*)gt)[tid];
    const float dx0 = p.x - g.x, dy0 = p.y - g.y;
    const float dx1 = p.z - g.z, dy1 = p.w - g.w;
    // safe norm: v_sqrt_f32(0) == 0, matching the torch-like zero subgrad.
    const float d0 = __builtin_amdgcn_sqrtf(dx0 * dx0 + dy0 * dy0);
    const float d1 = __builtin_amdgcn_sqrtf(dx1 * dx1 + dy1 * dy1);
    const float m0 = mask[b * AA + a]     ? 1.0f : 0.0f;
    const float m1 = mask[b * AA + a + 1] ? 1.0f : 0.0f;
    dsum = d0 * m0 + d1 * m1;
    fsum = (t == TT - 1) ? dsum : 0.0f;
  }
  // wave32 butterfly reduce
  for (int off = 16; off > 0; off >>= 1) {
    dsum += __shfl_down(dsum, off, 32);
    fsum += __shfl_down(fsum, off, 32);
  }
  if ((threadIdx.x & 31) == 0) {
    atomicAdd(&acc[2], dsum);
    atomicAdd(&acc[3], fsum);
  }
}

// ---------------------------------------------------------------------------
// Kernel 2: collision penalty. One 128-thread (4-wave) workgroup per (b,t).
// ---------------------------------------------------------------------------
__global__ void __launch_bounds__(128)
collision_kernel(const float* __restrict__ pred,
                 const unsigned char* __restrict__ mask,
                 float* __restrict__ acc) {
  __shared__ float2 sxy[AA];        // interleaved coords -> one b64 per agent
  __shared__ float  sn[AA];         // |x|^2 per agent (sentinel-aware)

  const int bt  = blockIdx.x;       // (b,t) slice, 0..1599
  const int b   = bt / TT;
  const int tid = threadIdx.x;      // 0..127

  // Stage coords into LDS; masked-out agents get distinct far-away sentinels
  // so every pair touching one fails dist < COLL_THR with huge margin
  // (sentinel separation >= 3e5; f32 noise in the Gram reconstruction is
  // orders of magnitude below d2 >= 9e10).
  const float2 p   = ((const float2*)pred)[bt * AA + tid];
  const bool   mv  = mask[b * AA + tid] != 0;
  const float  sen = 3.0e5f * (float)(tid + 1);
  const float  x   = mv ? p.x : sen;
  const float  y   = mv ? p.y : 0.0f;
  sxy[tid] = make_float2(x, y);
  sn[tid]  = x * x + y * y;
  __syncthreads();

  // Wave id as a true scalar: lane 0 of wave w has tid 32w.
  const int waveS  = __builtin_amdgcn_readfirstlane(threadIdx.x >> 5); // 0..3
  const int lane   = tid & 31;
  const int lane16 = lane & 15;
  const bool hi    = lane >= 16;    // lanes 16-31 carry K=2,3 (zero pad)

  float pen = 0.0f;
  float cnt = 0.0f;

  // 36 upper-triangular tiles (J >= I), 9 per wave, fully unrolled.
#pragma unroll
  for (int s = 0; s < 9; ++s) {
    const int ij = kTileIJ[waveS + 4 * s];   // scalar table lookup (s_load)
    const int I  = ij >> 4;
    const int J  = ij & 15;

    // Unconditional operand loads (addresses always valid), then cndmask
    // the K=2/3 zero padding — no exec-divergent loads around the WMMA.
    const float2 pa = sxy[I * 16 + lane16];
    const float2 pb = sxy[J * 16 + lane16];
    v2f av, bv;
    av[0] = hi ? 0.0f : pa.x;
    av[1] = hi ? 0.0f : pa.y;
    bv[0] = hi ? 0.0f : pb.x;
    bv[1] = hi ? 0.0f : pb.y;

    v8f c = {};
    // G[m][n] = x_{I*16+m} . x_{J*16+n}
    v8f g = __builtin_amdgcn_wmma_f32_16x16x4_f32(
        false, av, false, bv, (short)0, c, false, false);

    // Epilogue: C/D layout -> row = r (+8 in high lanes), col = lane16.
    const int   jg = J * 16 + lane16;
    const float nj = sn[jg];
    const int   ibase = I * 16 + (hi ? 8 : 0);
#pragma unroll
    for (int r = 0; r < 8; ++r) {
      const int   ig = ibase + r;
      const float ni = sn[ig];                 // merged into ds_load_b128 x2
      float d2 = ni + nj - 2.0f * g[r];
      d2 = (d2 > 0.0f) ? d2 : 0.0f;            // clamp cancellation noise
      const float dist = __builtin_amdgcn_sqrtf(d2);  // single v_sqrt_f32
      const bool hit = (ig < jg) & (dist < COLL_THR);
      pen += hit ? (COLL_THR - dist) : 0.0f;
      cnt += hit ? 1.0f : 0.0f;
    }
  }

  // wave32 reduce, then one atomic per wave.
  for (int off = 16; off > 0; off >>= 1) {
    pen += __shfl_down(pen, off, 32);
    cnt += __shfl_down(cnt, off, 32);
  }
  if (lane == 0) {
    atomicAdd(&acc[0], pen);
    atomicAdd(&acc[1], cnt);
  }
}

// ---------------------------------------------------------------------------
// Kernel 3: finalize. Single block: reduce mask sum, combine scalars.
// ---------------------------------------------------------------------------
__global__ void finalize_kernel(const unsigned char* __restrict__ mask,
                                const float* __restrict__ acc,
                                float* __restrict__ out) {
  __shared__ float red[256];
  float s = 0.0f;
  for (int i = threadIdx.x; i < BB * AA; i += 256)
    s += mask[i] ? 1.0f : 0.0f;
  red[threadIdx.x] = s;
  __syncthreads();
  for (int o = 128; o > 0; o >>= 1) {
    if (threadIdx.x < o) red[threadIdx.x] += red[threadIdx.x + o];
    __syncthreads();
  }
  if (threadIdx.x == 0) {
    const float maskSum = red[0];
    const float ade  = acc[2] / (maskSum * (float)TT);
    const float fde  = acc[3] / maskSum;
    const float coll = acc[0] / (acc[1] + 1e-8f);
    const float loss = ADE_W * ade + FDE_W * fde + COLL_W * coll;
    out[0] = loss;
    out[1] = ade;
    out[2] = fde;
    out[3] = coll;
  }
}

// ---------------------------------------------------------------------------
extern "C" void kernel_launch(void* const* d_in, const int* in_sizes, int n_in,
                              void* d_out, int out_size, void* d_ws, size_t ws_size,
                              hipStream_t stream) {
  const float* pred         = (const float*)d_in[0];          // (32,50,128,2) f32
  const float* gt           = (const float*)d_in[1];          // (32,50,128,2) f32
  const unsigned char* mask = (const unsigned char*)d_in[2];  // (32,128) bool (1B/elem)
  float* acc = (float*)d_ws;
  float* out = (float*)d_out;                                 // [loss, ade, fde, coll]

  hipMemsetAsync(d_ws, 0, 4 * sizeof(float), stream);         // graph-capturable

  const int n2 = BB * TT * AA / 2;                            // 102400 threads
  ade_fde_kernel<<<(n2 + 255) / 256, 256, 0, stream>>>(pred, gt, mask, acc);
  collision_kernel<<<BB * TT, 128, 0, stream>>>(pred, mask, acc);
  finalize_kernel<<<1, 256, 0, stream>>>(mask, acc, out);
}